// GCN_57002805952623
// MI455X (gfx1250) — compile-verified
//
#include <hip/hip_runtime.h>

#define D_IN  512
#define HID   1024
#define D_OUT 512

typedef float v2f __attribute__((ext_vector_type(2)));
typedef float v8f __attribute__((ext_vector_type(8)));

// ---------------- degree / normalization ----------------

__global__ void k_deg_init(float* __restrict__ deg, int N) {
    int i = blockIdx.x * blockDim.x + threadIdx.x;
    if (i < N) deg[i] = 1.0f;   // self-loop contributes 1 to in-degree
}

__global__ void k_deg_accum(float* __restrict__ deg, const int* __restrict__ dst, int E) {
    int e = blockIdx.x * blockDim.x + threadIdx.x;
    if (e < E) atomicAdd(&deg[dst[e]], 1.0f);
}

__global__ void k_dinv(float* __restrict__ deg, int N) {
    int i = blockIdx.x * blockDim.x + threadIdx.x;
    if (i < N) {
        float d = deg[i];
        deg[i] = (d > 0.0f) ? rsqrtf(d) : 0.0f;   // in-place deg -> dinv
    }
}

// ---------------- f32 WMMA GEMM: C[Nrows,M] = A[Nrows,K] @ B[K,M] ----------------
// Block = 256 threads = 8 waves (wave32), arranged 2 wave-rows x 4 wave-cols.
// Block tile 128x128. Wave tile 64x32 = 4x2 subtiles of 16x16 -> per k-step
// 8 VMEM : 8 WMMA. K and M are compile-time so all strides fold into
// immediate offsets; interior waves take a guard-free path (wave-uniform
// branch, EXEC stays all-ones around the WMMAs).
// A-frag (16x4): lanes 0-15 hold K=k0,k0+1 ; lanes 16-31 hold K=k0+2,k0+3.
// B-frag (4x16): same K striping, lane = column.
// C/D (16x16): VGPR v: lanes 0-15 -> M=v, lanes 16-31 -> M=v+8.

template <int K, int M, bool GUARD>
__device__ __forceinline__
void gemm_body(const float* __restrict__ A, const float* __restrict__ B,
               float* __restrict__ C, int Nrows,
               int row0, int col0, int half, int l16) {
    v8f c[4][2] = {};

    for (int k0 = 0; k0 < K; k0 += 4) {
        v2f a[4], b[2];
        #pragma unroll
        for (int i = 0; i < 4; ++i) {
            int ar = row0 + 16 * i + l16;
            if (GUARD) {
                int arc = (ar < Nrows) ? ar : (Nrows - 1);   // clamp address, zero value
                const float* ap = A + (size_t)arc * K + k0 + 2 * half;
                v2f av; av.x = ap[0]; av.y = ap[1];
                if (ar >= Nrows) { av.x = 0.0f; av.y = 0.0f; }
                a[i] = av;
            } else {
                const float* ap = A + (size_t)ar * K + k0 + 2 * half;
                v2f av; av.x = ap[0]; av.y = ap[1];          // contiguous -> b64 load
                a[i] = av;
            }
        }
        #pragma unroll
        for (int j = 0; j < 2; ++j) {
            const float* bp = B + (size_t)(k0 + 2 * half) * M + col0 + 16 * j + l16;
            v2f bv; bv.x = bp[0]; bv.y = bp[M];              // second row = imm offset M*4
            b[j] = bv;
        }
        #pragma unroll
        for (int i = 0; i < 4; ++i)
            #pragma unroll
            for (int j = 0; j < 2; ++j)
                c[i][j] = __builtin_amdgcn_wmma_f32_16x16x4_f32(
                    /*neg_a=*/false, a[i], /*neg_b=*/false, b[j],
                    /*c_mod=*/(short)0, c[i][j], /*reuse_a=*/false, /*reuse_b=*/false);
    }

    #pragma unroll
    for (int i = 0; i < 4; ++i) {
        #pragma unroll
        for (int j = 0; j < 2; ++j) {
            int rbase = row0 + 16 * i + 8 * half;
            float* p = C + (size_t)rbase * M + col0 + 16 * j + l16;
            #pragma unroll
            for (int v = 0; v < 8; ++v) {
                if (!GUARD || (rbase + v) < Nrows)
                    p[(size_t)v * M] = c[i][j][v];           // imm offsets v*M*4
            }
        }
    }
}

template <int K, int M>
__global__ __launch_bounds__(256)
void k_gemm_wmma_f32(const float* __restrict__ A, const float* __restrict__ B,
                     float* __restrict__ C, int Nrows) {
    const int lane = threadIdx.x & 31;
    const int wave = threadIdx.x >> 5;
    const int wm   = wave >> 2;          // 0..1
    const int wn   = wave & 3;           // 0..3
    const int half = lane >> 4;          // 0 or 1
    const int l16  = lane & 15;

    const int row0 = blockIdx.y * 128 + wm * 64;
    const int col0 = blockIdx.x * 128 + wn * 32;

    if (row0 + 64 <= Nrows)              // wave-uniform: fast interior path
        gemm_body<K, M, false>(A, B, C, Nrows, row0, col0, half, l16);
    else
        gemm_body<K, M, true >(A, B, C, Nrows, row0, col0, half, l16);
}

// ---------------- aggregation ----------------

// acc[i,:] = dinv[i]^2 * xw[i,:]   (self-loop term), float4-vectorized
__global__ void k_agg_init(float4* __restrict__ acc, const float4* __restrict__ xw,
                           const float* __restrict__ dinv, int N, int Fv) {
    long long t = (long long)blockIdx.x * blockDim.x + threadIdx.x;
    if (t < (long long)N * Fv) {
        int i = (int)(t / Fv);
        float s = dinv[i]; s *= s;
        float4 v = xw[t];
        float4 r; r.x = v.x * s; r.y = v.y * s; r.z = v.z * s; r.w = v.w * s;
        acc[t] = r;
    }
}

// one block per edge: acc[dst,:] += dinv[src]*dinv[dst] * xw[src,:]
__global__ void k_agg_edges(float* __restrict__ acc, const float* __restrict__ xw,
                            const float* __restrict__ dinv,
                            const int* __restrict__ src, const int* __restrict__ dst,
                            int F) {
    int e = blockIdx.x;
    int s = src[e], d = dst[e];
    float nrm = dinv[s] * dinv[d];
    const float4* xs = (const float4*)(xw + (size_t)s * F);
    float* ad = acc + (size_t)d * F;
    for (int f4 = threadIdx.x; f4 * 4 < F; f4 += blockDim.x) {
        float4 v = xs[f4];
        int f = f4 * 4;
        atomicAdd(ad + f + 0, v.x * nrm);
        atomicAdd(ad + f + 1, v.y * nrm);
        atomicAdd(ad + f + 2, v.z * nrm);
        atomicAdd(ad + f + 3, v.w * nrm);
    }
}

// h = relu(h + b) elementwise, in place
__global__ void k_bias_relu(float* __restrict__ h, const float* __restrict__ b,
                            long long total, int F) {
    long long t = (long long)blockIdx.x * blockDim.x + threadIdx.x;
    if (t < total) {
        int f = (int)(t % F);
        float v = h[t] + b[f];
        h[t] = (v > 0.0f) ? v : 0.0f;
    }
}

// out[f, i] = acc[i, f] + bias[f]   (LDS-tiled transpose, both sides coalesced)
__global__ __launch_bounds__(256)
void k_transpose_bias(float* __restrict__ out, const float* __restrict__ acc,
                      const float* __restrict__ bias, int N, int F) {
    __shared__ float tile[32][33];
    int i0 = blockIdx.x * 32;
    int f0 = blockIdx.y * 32;
    int tx = threadIdx.x, ty = threadIdx.y;   // block (32, 8)
    #pragma unroll
    for (int r = 0; r < 4; ++r) {
        int i = i0 + ty + 8 * r;
        if (i < N) tile[ty + 8 * r][tx] = acc[(size_t)i * F + f0 + tx];
    }
    __syncthreads();
    #pragma unroll
    for (int r = 0; r < 4; ++r) {
        int f = f0 + ty + 8 * r;
        int i = i0 + tx;
        if (i < N) out[(size_t)f * N + i] = tile[tx][ty + 8 * r] + bias[f];
    }
}

// ---------------- driver ----------------

extern "C" void kernel_launch(void* const* d_in, const int* in_sizes, int n_in,
                              void* d_out, int out_size, void* d_ws, size_t ws_size,
                              hipStream_t stream) {
    const float* x  = (const float*)d_in[0];
    const float* W1 = (const float*)d_in[1];
    const float* b1 = (const float*)d_in[2];
    const float* W2 = (const float*)d_in[3];
    const float* b2 = (const float*)d_in[4];
    const int*   ei = (const int*)d_in[5];

    const int N = in_sizes[0] / D_IN;
    const int E = in_sizes[5] / 2;
    const int* srcp = ei;
    const int* dstp = ei + E;

    char* w = (char*)d_ws;
    size_t off = 0;
    auto alloc = [&](size_t bytes) -> void* {
        void* p = w + off;
        off = (off + bytes + 255) & ~(size_t)255;
        return p;
    };

    float* dinv = (float*)alloc((size_t)N * 4);
    float* xw1  = (float*)alloc((size_t)N * HID * 4);
    float* acc1 = (float*)alloc((size_t)N * HID * 4);   // becomes H after relu
    float* hw2  = (float*)alloc((size_t)N * D_OUT * 4);
    float* acc2 = xw1;                                  // reuse: xw1 dead after layer-1 agg

    // 1. symmetric-normalization coefficients
    k_deg_init <<<(N + 255) / 256, 256, 0, stream>>>(dinv, N);
    k_deg_accum<<<(E + 255) / 256, 256, 0, stream>>>(dinv, dstp, E);
    k_dinv     <<<(N + 255) / 256, 256, 0, stream>>>(dinv, N);

    // 2. XW1 = X @ W1   [N,512]@[512,1024]
    {
        dim3 grid(HID / 128, (N + 127) / 128);
        k_gemm_wmma_f32<D_IN, HID><<<grid, 256, 0, stream>>>(x, W1, xw1, N);
    }

    // 3. layer-1 aggregation + bias + relu
    {
        long long tot4 = (long long)N * (HID / 4);
        k_agg_init<<<(unsigned)((tot4 + 255) / 256), 256, 0, stream>>>(
            (float4*)acc1, (const float4*)xw1, dinv, N, HID / 4);
        k_agg_edges<<<E, 256, 0, stream>>>(acc1, xw1, dinv, srcp, dstp, HID);
        long long tot = (long long)N * HID;
        k_bias_relu<<<(unsigned)((tot + 255) / 256), 256, 0, stream>>>(acc1, b1, tot, HID);
    }

    // 4. HW2 = H @ W2   [N,1024]@[1024,512]
    {
        dim3 grid(D_OUT / 128, (N + 127) / 128);
        k_gemm_wmma_f32<HID, D_OUT><<<grid, 256, 0, stream>>>(acc1, W2, hw2, N);
    }

    // 5. layer-2 aggregation
    {
        long long tot4 = (long long)N * (D_OUT / 4);
        k_agg_init<<<(unsigned)((tot4 + 255) / 256), 256, 0, stream>>>(
            (float4*)acc2, (const float4*)hw2, dinv, N, D_OUT / 4);
        k_agg_edges<<<E, 128, 0, stream>>>(acc2, hw2, dinv, srcp, dstp, D_OUT);
    }

    // 6. out[f, i] = acc2[i, f] + b2[f]
    {
        dim3 grid((N + 31) / 32, D_OUT / 32);
        dim3 blk(32, 8);
        k_transpose_bias<<<grid, blk, 0, stream>>>((float*)d_out, acc2, b2, N, D_OUT);
    }
}